// KMeans_57492432224463
// MI455X (gfx1250) — compile-verified
//
#include <hip/hip_runtime.h>

typedef float v2f __attribute__((ext_vector_type(2)));
typedef float v8f __attribute__((ext_vector_type(8)));

#define D_DIM   393216
#define N_PTS   512
#define K_CL    10
#define NTILE   16                       // centroid count padded to WMMA N
#define CHUNK   8192                     // K-elements per block
#define NCHUNK  (D_DIM / CHUNK)          // 48
#define WPB     4                        // waves per block
#define WAVE_K  (CHUNK / WPB)            // 2048 K-elements per wave

// ---------------- workspace zeroing ----------------
__global__ void KMeans_zero_kernel(float* __restrict__ ws, int n) {
    int i = blockIdx.x * blockDim.x + threadIdx.x;
    if (i < n) ws[i] = 0.0f;
}

// ---------------- ||c_j||^2 ----------------
// grid: K_CL * 16 blocks of 256 threads; block handles 1/16 of one centroid row
__global__ void KMeans_c2_kernel(const float* __restrict__ c, float* __restrict__ c2) {
    const int j     = blockIdx.x >> 4;
    const int chunk = blockIdx.x & 15;
    const int len   = D_DIM / 16;                       // 24576
    const float* p  = c + (size_t)j * D_DIM + (size_t)chunk * len;
    float s = 0.0f;
    for (int i = threadIdx.x; i < len; i += 256) {
        float v = p[i];
        s += v * v;
    }
    __shared__ float red[256];
    red[threadIdx.x] = s;
    __syncthreads();
    for (int off = 128; off > 0; off >>= 1) {
        if (threadIdx.x < off) red[threadIdx.x] += red[threadIdx.x + off];
        __syncthreads();
    }
    if (threadIdx.x == 0) atomicAdd(&c2[j], red[0]);
}

// ---------------- fp32 WMMA GEMM: dot[i][j] += x_i . c_j ----------------
// grid: (32 M-tiles, 48 K-chunks), block = 128 threads (4 waves).
// Padded B columns 10..15 use a clamped (duplicated) centroid row: their dot
// products are garbage but never read by the argmin kernel, so no predication
// or zero-fill is needed in the hot loop (keeps EXEC all-1s, no divergence).
__global__ void KMeans_dot_kernel(const float* __restrict__ x,
                                  const float* __restrict__ c,
                                  float* __restrict__ dotAcc) {
    const int mtile  = blockIdx.x;            // 0..31
    const int kchunk = blockIdx.y;            // 0..47
    const int wave   = threadIdx.x >> 5;      // 0..3
    const int lane   = threadIdx.x & 31;

    const int  mrow  = lane & 15;             // row within tile (both half-waves)
    const int  ncol  = lane & 15;             // B/N column within tile
    const int  koff  = (lane >> 4) << 1;      // 0 for lanes 0-15, 2 for lanes 16-31
    const long kbase = (long)kchunk * CHUNK + (long)wave * WAVE_K;

    const int  crow  = (ncol < K_CL) ? ncol : (ncol - 8);  // clamp pad cols to valid rows
    const float* xp  = x + (size_t)(mtile * 16 + mrow) * D_DIM + kbase + koff;
    const float* cp  = c + (size_t)crow * D_DIM + kbase + koff;

    v8f acc = {};
    for (int k = 0; k < WAVE_K; k += 16) {
        // A tile: 16x16 of x as 4 (16x4) fragments; float2 per lane, NT (x is single-use)
        v2f a0 = __builtin_nontemporal_load((const v2f*)(xp + k));
        v2f a1 = __builtin_nontemporal_load((const v2f*)(xp + k + 4));
        v2f a2 = __builtin_nontemporal_load((const v2f*)(xp + k + 8));
        v2f a3 = __builtin_nontemporal_load((const v2f*)(xp + k + 12));
        // B tile: centroids^T (L2-resident, regular temporal loads)
        v2f b0 = *(const v2f*)(cp + k);
        v2f b1 = *(const v2f*)(cp + k + 4);
        v2f b2 = *(const v2f*)(cp + k + 8);
        v2f b3 = *(const v2f*)(cp + k + 12);
        acc = __builtin_amdgcn_wmma_f32_16x16x4_f32(false, a0, false, b0, (short)0, acc, false, false);
        acc = __builtin_amdgcn_wmma_f32_16x16x4_f32(false, a1, false, b1, (short)0, acc, false, false);
        acc = __builtin_amdgcn_wmma_f32_16x16x4_f32(false, a2, false, b2, (short)0, acc, false, false);
        acc = __builtin_amdgcn_wmma_f32_16x16x4_f32(false, a3, false, b3, (short)0, acc, false, false);
    }

    // Reduce the 4 waves' partial 16x16 tiles in LDS, then one atomic per cell to L2
    __shared__ float tile[256];
    for (int i = threadIdx.x; i < 256; i += blockDim.x) tile[i] = 0.0f;
    __syncthreads();

    // D layout: VGPR r -> (M=r, N=lane) for lanes 0-15, (M=r+8, N=lane-16) for 16-31
    const int mbase = (lane >> 4) * 8;
#pragma unroll
    for (int r = 0; r < 8; ++r) {
        atomicAdd(&tile[(mbase + r) * 16 + ncol], acc[r]);
    }
    __syncthreads();

    for (int i = threadIdx.x; i < 256; i += blockDim.x) {
        int m = i >> 4, n = i & 15;
        atomicAdd(&dotAcc[(size_t)(mtile * 16 + m) * NTILE + n], tile[i]);
    }
}

// ---------------- argmin over j of (||c_j||^2 - 2 dot_ij) ----------------
// (||x_i||^2 is row-constant and cannot change the argmin, so it is dropped.)
__global__ void KMeans_argmin_kernel(const float* __restrict__ c2,
                                     const float* __restrict__ dotAcc,
                                     int* __restrict__ out) {
    int i = blockIdx.x * blockDim.x + threadIdx.x;
    if (i >= N_PTS) return;
    float best = 3.4e38f;
    int   bj   = 0;
#pragma unroll
    for (int j = 0; j < K_CL; ++j) {
        float s = c2[j] - 2.0f * dotAcc[(size_t)i * NTILE + j];
        if (s < best) { best = s; bj = j; }
    }
    out[i] = bj;
}

extern "C" void kernel_launch(void* const* d_in, const int* in_sizes, int n_in,
                              void* d_out, int out_size, void* d_ws, size_t ws_size,
                              hipStream_t stream) {
    const float* x = (const float*)d_in[0];       // [512, 393216]
    const float* c = (const float*)d_in[1];       // [10, 393216]

    float* ws     = (float*)d_ws;
    float* c2     = ws;                            // 16 floats (10 used)
    float* dotAcc = ws + NTILE;                    // 512*16 floats
    int*   out    = (int*)d_out;                   // 512 int32 cluster ids

    const int ws_elems = NTILE + N_PTS * NTILE;    // 8208
    KMeans_zero_kernel<<<(ws_elems + 255) / 256, 256, 0, stream>>>(ws, ws_elems);

    KMeans_c2_kernel<<<K_CL * 16, 256, 0, stream>>>(c, c2);

    dim3 grid(N_PTS / 16, NCHUNK);                 // 32 x 48
    KMeans_dot_kernel<<<grid, WPB * 32, 0, stream>>>(x, c, dotAcc);

    KMeans_argmin_kernel<<<(N_PTS + 255) / 256, 256, 0, stream>>>(c2, dotAcc, out);
}